// ConvCaps_15083925144073
// MI455X (gfx1250) — compile-verified
//
#include <hip/hip_runtime.h>
#include <math.h>

// ---------------------------------------------------------------------------
// ConvCaps EM-routing, fused for MI455X (gfx1250, wave32).
//   prep_kernel : S (pooled x), AC (pooled a), WS (q'-summed w) -> workspace
//   em_kernel   : 1 workgroup per routing group (2304 groups);
//                 WS rows staged LDS via global_load_async_to_lds_b128;
//                 vv (16KB) lives in LDS; 3 EM iterations fused;
//                 32-deep column reductions via v_wmma_f32_16x16x4_f32
//                 (ones-matrix A trick), uniform scalar-branch structure.
// ---------------------------------------------------------------------------

typedef __attribute__((ext_vector_type(2))) float v2f;
typedef __attribute__((ext_vector_type(8))) float v8f;

#define EPSF      1e-8f
#define LOG2PIF   1.8378770664093453f

// workspace layout (floats)
#define S_OFF     0          // 1024*36   = 36864
#define AC_OFF    36864      // 64*36     = 2304
#define WS_OFF    39168      // 1152*128  = 147456
#define PREP_TOTAL 186624

#define NGROUP    2304       // 64 * 36
#define MU_ELEMS  1179648    // 2304 * 512

#define WMMA_F32X4(acc, A, B) \
  __builtin_amdgcn_wmma_f32_16x16x4_f32(false, (A), false, (B), (short)0, (acc), false, false)

// ---------------------------------------------------------------------------
__global__ __launch_bounds__(256) void prep_kernel(const float* __restrict__ x,
                                                   const float* __restrict__ a,
                                                   const float* __restrict__ w,
                                                   float* __restrict__ ws) {
  int idx = blockIdx.x * blockDim.x + threadIdx.x;
  if (idx >= PREP_TOTAL) return;

  if (idx < 36864) {                       // S: pooled x view (1024,32,14,14)
    int m2 = idx / 36, rem = idx % 36, V = rem / 6, U = rem % 6;
    const float* p = x + m2 * 6272 + (2 * V) * 14 + 2 * U;
    float s = 0.f;
#pragma unroll 4
    for (int c = 0; c < 32; ++c) {
      const float* q = p + c * 196;
      s += q[0] + q[1] + q[2] + q[14] + q[15] + q[16] + q[28] + q[29] + q[30];
    }
    ws[S_OFF + idx] = s * (1.0f / 9.0f);
  } else if (idx < 39168) {                // AC: pooled a (64,32,14,14)
    int j = idx - 36864;
    int b = j / 36, rem = j % 36, Y = rem / 6, X = rem % 6;
    const float* p = a + b * 6272 + (2 * Y) * 14 + 2 * X;
    float s = 0.f;
#pragma unroll 4
    for (int c = 0; c < 32; ++c) {
      const float* q = p + c * 196;
      s += q[0] + q[1] + q[2] + q[14] + q[15] + q[16] + q[28] + q[29] + q[30];
    }
    ws[AC_OFF + j] = s * (1.0f / 9.0f);
  } else {                                 // WS: q'-reduced voting weights
    int j = idx - 39168;                   // j = s2*128 + O*4 + r
    int s2 = j >> 7, Or = j & 127, O = Or >> 2, r = Or & 3;
    int base = s2 * 512 + (O >> 3) * 128 + ((O >> 1) & 3) * 32 + (O & 1) * 16 + r;
    ws[WS_OFF + j] = w[base] + w[base + 4] + w[base + 8] + w[base + 12];
  }
}

// ---------------------------------------------------------------------------
__global__ __launch_bounds__(256) void em_kernel(const float* __restrict__ ws,
                                                 const float* __restrict__ b_u,
                                                 const float* __restrict__ b_a,
                                                 float* __restrict__ out_mu,
                                                 float* __restrict__ out_a) {
  __shared__ float wstage[32][128];  // WS rows, filled by async-to-LDS b128
  __shared__ float vv[32][132];      // vv[m][O*4+r] (pp compacted to 4), padded
  __shared__ float rmat[32][33];     // routing weights r, padded
  __shared__ float Rb[32][33];       // r*a + eps, padded
  __shared__ float ap_s[32][33];     // e-step logits, padded
  __shared__ float avv[32];
  __shared__ float colsum[288];      // [0,128): ΣRv  [128,256): ΣRv²  [256,288): r_sum
  __shared__ float mu_s[32][4];
  __shared__ float sig_s[32][4];
  __shared__ float rsum_s[32];
  __shared__ float aout_s[32];
  __shared__ float bu_s[32], ba_s[32];

  const int g = blockIdx.x;           // 0..2303
  const int b = g / 36, t = g % 36;
  const int tid  = threadIdx.x;
  const int lane = tid & 31;
  const int wave = __builtin_amdgcn_readfirstlane(tid >> 5);  // scalar wave id

  const float* S  = ws + S_OFF;
  const float* AC = ws + AC_OFF;
  const float* WS = ws + WS_OFF;

  // ---- async-stage this wave's 4 WS rows (512B each) into LDS ------------
#pragma unroll
  for (int k = 0; k < 4; ++k) {
    int m = wave * 4 + k;
    int sm = t * 32 + m;
    unsigned lds  = (unsigned)(uintptr_t)&wstage[m][lane * 4];
    unsigned voff = (unsigned)(sm * 512 + lane * 16);   // byte offset into WS
    asm volatile("global_load_async_to_lds_b128 %0, %1, %2"
                 :: "v"(lds), "v"(voff), "s"(WS) : "memory");
  }

  // ---- scalars + r init (overlaps the async copies) ----------------------
  if (tid < 32) {
    bu_s[tid] = b_u[tid];
    ba_s[tid] = b_a[tid];
    int sm = t * 32 + tid;
    avv[tid] = AC[b * 36 + (sm % 36)];
  }
  {
    int m = tid >> 3, ob = (tid & 7) * 4;
    for (int o = ob; o < ob + 4; ++o) rmat[m][o] = 1.0f / 32.0f;
  }

  asm volatile("s_wait_asynccnt 0" ::: "memory");

  // ---- vv[m][O*4+r] = XC(m,O) * wstage[m][O*4+r] -------------------------
  {
    int m = tid >> 3;
    int sm = t * 32 + m;
    int i = sm / 36, pos = sm % 36;
    int Y = pos / 6, X = pos % 6;
    int m2 = b * 16 + (i >> 1);
    int r0base = (i & 1) * 576 + Y * 96 + X * 16;
    int Ob = (tid & 7) * 4;
    for (int O = Ob; O < Ob + 4; ++O) {
      int pq = ((O >> 3) << 2) | ((O >> 1) & 3);
      float xc = S[m2 * 36 + ((r0base + pq) % 36)];
      for (int r = 0; r < 4; ++r) vv[m][O * 4 + r] = xc * wstage[m][O * 4 + r];
    }
  }
  __syncthreads();

  const float lambds[3] = {0.0005f, 0.000975f, 0.00142625f};  // 0.01*(1-0.95^(it+1))
  v2f ones; ones.x = 1.0f; ones.y = 1.0f;

  for (int it = 0; it < 3; ++it) {
    // ---- m-step: Rb = r*a + eps ------------------------------------------
    {
      int m = tid >> 3, ob = (tid & 7) * 4;
      float av = avv[m];
      for (int o = ob; o < ob + 4; ++o) Rb[m][o] = rmat[m][o] * av + EPSF;
    }
    __syncthreads();

    // ---- 32-deep column reductions via WMMA (ones-A), uniform sections ---
    {
      const int cidx = (wave << 4) + (lane & 15);  // column o*4+r, 0..127
      const int oidx = cidx >> 2;
      const int kb   = (lane >> 4) << 1;           // lanes 0-15: K rows 0,1 ; 16-31: 2,3

      v8f acc = {};                                // ΣRv (tiles 0..7)
#pragma unroll
      for (int c = 0; c < 8; ++c) {
        int mr = c * 4 + kb;
        v2f Bv;
        Bv.x = Rb[mr][oidx]     * vv[mr][cidx];
        Bv.y = Rb[mr + 1][oidx] * vv[mr + 1][cidx];
        acc = WMMA_F32X4(acc, ones, Bv);
      }
      if (lane < 16) colsum[(wave << 4) + lane] = acc[0];

      v8f acc2 = {};                               // ΣRv² (tiles 8..15)
#pragma unroll
      for (int c = 0; c < 8; ++c) {
        int mr = c * 4 + kb;
        float v0 = vv[mr][cidx], v1 = vv[mr + 1][cidx];
        v2f Bv;
        Bv.x = Rb[mr][oidx]     * v0 * v0;
        Bv.y = Rb[mr + 1][oidx] * v1 * v1;
        acc2 = WMMA_F32X4(acc2, ones, Bv);
      }
      if (lane < 16) colsum[128 + (wave << 4) + lane] = acc2[0];

      if (wave < 2) {                              // ΣRb = r_sum (tiles 16,17)
        int o = (wave << 4) + (lane & 15);
        v8f acc3 = {};
#pragma unroll
        for (int c = 0; c < 8; ++c) {
          int mr = c * 4 + kb;
          v2f Bv;
          Bv.x = Rb[mr][o];
          Bv.y = Rb[mr + 1][o];
          acc3 = WMMA_F32X4(acc3, ones, Bv);
        }
        if (lane < 16) colsum[256 + (wave << 4) + lane] = acc3[0];
      }
    }
    __syncthreads();

    // ---- stats: mu, sig_sq, r_sum ----------------------------------------
    if (tid < 128) {
      int o = tid >> 2, r = tid & 3;
      float rs = colsum[256 + o];
      float mu = colsum[o * 4 + r] / rs;
      float sg = colsum[128 + o * 4 + r] / rs - mu * mu + EPSF;
      mu_s[o][r]  = mu;
      sig_s[o][r] = sg;
      if (r == 0) rsum_s[o] = rs;
    }
    __syncthreads();

    // ---- cost + activation -----------------------------------------------
    if (tid < 32) {
      int o = tid;
      float cs = 0.f;
      for (int r = 0; r < 4; ++r) cs += bu_s[o] + 0.5f * __logf(sig_s[o][r]);
      cs = cs * 4.0f * rsum_s[o];                 // Σ over 16 pp (period-4)
      float z = lambds[it] * (ba_s[o] - cs);
      aout_s[o] = 1.0f / (1.0f + __expf(-z));
    }
    __syncthreads();

    if (it < 2) {
      // ---- e-step: ap_ln then softmax over o -----------------------------
      {
        int m = tid >> 3, ob = (tid & 7) * 4;
        for (int o = ob; o < ob + 4; ++o) {
          float s = 0.f;
          for (int r = 0; r < 4; ++r) {
            float d  = vv[m][o * 4 + r] - mu_s[o][r];
            float sg = sig_s[o][r];
            s += -d * d / (2.0f * sg) - 0.5f * __logf(sg);
          }
          ap_s[m][o] = 4.0f * s - 8.0f * LOG2PIF + __logf(aout_s[o]);
        }
      }
      __syncthreads();
      if (tid < 32) {
        int m = tid;
        float mx = -1e30f;
        for (int o = 0; o < 32; ++o) mx = fmaxf(mx, ap_s[m][o]);
        float tot = 0.f;
        for (int o = 0; o < 32; ++o) tot += __expf(ap_s[m][o] - mx);
        float inv = 1.0f / tot;
        for (int o = 0; o < 32; ++o) rmat[m][o] = __expf(ap_s[m][o] - mx) * inv;
      }
      __syncthreads();
    }
  }

  // ---- outputs (contiguous per group by construction) ----------------------
  if (tid < 128) {
    int o = tid >> 2, r = tid & 3;
    float mv = mu_s[o][r];
    float* om = out_mu + (size_t)g * 512 + o * 16;
    om[r] = mv; om[r + 4] = mv; om[r + 8] = mv; om[r + 12] = mv;  // pp period-4
  }
  if (tid < 32) out_a[(size_t)g * 32 + tid] = aout_s[tid];
}

// ---------------------------------------------------------------------------
extern "C" void kernel_launch(void* const* d_in, const int* in_sizes, int n_in,
                              void* d_out, int out_size, void* d_ws, size_t ws_size,
                              hipStream_t stream) {
  const float* x   = (const float*)d_in[0];
  const float* a   = (const float*)d_in[1];
  const float* w   = (const float*)d_in[2];
  const float* b_u = (const float*)d_in[3];
  const float* b_a = (const float*)d_in[4];
  float* ws  = (float*)d_ws;
  float* out = (float*)d_out;

  prep_kernel<<<PREP_TOTAL / 256, 256, 0, stream>>>(x, a, w, ws);
  em_kernel<<<NGROUP, 256, 0, stream>>>(ws, b_u, b_a, out, out + MU_ELEMS);
}